// PulseRetrievalLossFunctionHilbertFrog_30897994727536
// MI455X (gfx1250) — compile-verified
//
#include <hip/hip_runtime.h>
#include <hip/hip_bf16.h>

// ============================================================================
// PulseRetrievalLossFunctionHilbertFrog for MI455X (gfx1250)
//
// One workgroup (256 threads = 8 waves, wave32) per batch row.
// LDS: two complex ping-pong buffers (4 x 8192 floats = 128 KB dynamic LDS;
// WGP has 320 KB so 2 workgroups/WGP overlap memory and compute across rows).
//
// 8192-pt FFT = 3 x radix-16 Stockham DIF stages (16x16x16 complex matmuls on
// V_WMMA_F32_16X16X4_F32, K accumulated in 4 chunks of 4) + one radix-2 stage.
// Stockham is self-sorting -> natural-order output.
//
// WMMA fp32 layout assumptions (CDNA5 ISA):
//   A (16x4): 2 VGPRs; lane L: M=L&15, holds K = {0,1} (L<16) or {2,3} (L>=16)
//             within the chunk (a.x = even K, a.y = odd K).
//   B (4x16): mirrored: lane L: N=L&15, b.x = B[Keven][N], b.y = B[Kodd][N],
//             Keven = 2*(L>=16) within the chunk.
//   C/D (16x16): 8 VGPRs; lane L, elem e: M = e + 8*(L>=16), N = L&15.
//
// MI455X data-path choices:
//  - fp32 WMMA (not f16/fp8): an FFT + squared-error loss needs fp32 precision;
//    V_WMMA_F32_16X16X4_F32 is the CDNA5 op built for exactly this.
//  - global_prefetch_b8 of the label row at kernel start: labels are 32 MiB
//    total and the 192 MB device L2 holds them all, so the end-of-chain scan
//    pass hits L2 instead of exposed HBM latency.
//  - stage-0 scatter uses ds_store_b128 (consecutive output indices per lane).
// ============================================================================

typedef float v2f __attribute__((ext_vector_type(2)));
typedef float v8f __attribute__((ext_vector_type(8)));

#define FFT_N   8192
#define BATCH   512
#define TPB     256

__constant__ float kPI2 = 6.283185307179586f;

__device__ __forceinline__ v8f wmma4_f32(v2f a, v2f b, v8f c) {
    // 8 args: (neg_a, A, neg_b, B, c_mod, C, reuse_a, reuse_b)
    return __builtin_amdgcn_wmma_f32_16x16x4_f32(false, a, false, b,
                                                 (short)0, c, false, false);
}

struct Dft16Frag {
    v2f cosv[4];   // A_re = cos(-2*pi*m*k/16)
    v2f sinP[4];   // A_im = sin(-2*pi*m*k/16)
    v2f sinN[4];   // -A_im  (for D_re accumulation; NEG bits illegal on f32)
};

// One radix-16 DIF Stockham stage over the whole 8192-pt array.
// All three stages gather B[j][c] = x[16u + c + 512 j] (u = tile 0..31);
// only twiddle angle and output scatter depend on the stage.
template <int STAGE>
__device__ __forceinline__ void radix16_wmma_stage(
    const float* __restrict__ xr, const float* __restrict__ xi,
    float* __restrict__ yr, float* __restrict__ yi,
    int wave, int hi, int nc, const Dft16Frag& A)
{
    for (int t = 0; t < 4; ++t) {
        const int u    = wave * 4 + t;       // tile index 0..31 (uniform/wave)
        const int base = 16 * u + nc;

        v8f cre = {0.f,0.f,0.f,0.f,0.f,0.f,0.f,0.f};
        v8f cim = {0.f,0.f,0.f,0.f,0.f,0.f,0.f,0.f};
#pragma unroll
        for (int kk = 0; kk < 4; ++kk) {
            const int k0 = 4 * kk + 2 * hi;  // this lane's even K row
            const int a0 = base + 512 * k0;
            v2f bre, bim;
            bre.x = xr[a0];       bre.y = xr[a0 + 512];
            bim.x = xi[a0];       bim.y = xi[a0 + 512];
            // D_re += Are*Bre + (-Aim)*Bim ; D_im += Aim*Bre + Are*Bim
            cre = wmma4_f32(A.cosv[kk], bre, cre);
            cre = wmma4_f32(A.sinN[kk], bim, cre);
            cim = wmma4_f32(A.sinP[kk], bre, cim);
            cim = wmma4_f32(A.cosv[kk], bim, cim);
        }

        // Twiddle (e^{-2*pi*i*p*k/n}) + stage-specific scatter.
        float outR[8], outI[8];
        int   idxA[8];
#pragma unroll
        for (int e = 0; e < 8; ++e) {
            const int m = e + 8 * hi;        // DFT16 output index k
            float ang;
            if (STAGE == 0) {                // n=8192, s=1:  p = 16u+nc
                idxA[e] = 256 * u + 16 * nc + m;
                ang = (float)((16 * u + nc) * m) * (-kPI2 / 8192.0f);
            } else if (STAGE == 1) {         // n=512, s=16:  p = u, q = nc
                idxA[e] = 256 * u + nc + 16 * m;
                ang = (float)(u * m) * (-kPI2 / 512.0f);
            } else {                         // n=32, s=256:  p = u>>4
                const int p  = u >> 4;
                const int tt = u & 15;
                idxA[e] = 16 * tt + nc + 4096 * p + 256 * m;
                ang = (float)(p * m) * (-kPI2 / 32.0f);
            }
            float sn, cs;
            __sincosf(ang, &sn, &cs);
            const float dr = cre[e], di = cim[e];
            outR[e] = dr * cs - di * sn;
            outI[e] = dr * sn + di * cs;
        }

        if (STAGE == 0) {
            // Consecutive output indices per lane -> two ds_store_b128 each.
            const int bi = 256 * u + 16 * nc + 8 * hi;   // 32B-aligned
            *(float4*)&yr[bi]     = make_float4(outR[0], outR[1], outR[2], outR[3]);
            *(float4*)&yr[bi + 4] = make_float4(outR[4], outR[5], outR[6], outR[7]);
            *(float4*)&yi[bi]     = make_float4(outI[0], outI[1], outI[2], outI[3]);
            *(float4*)&yi[bi + 4] = make_float4(outI[4], outI[5], outI[6], outI[7]);
        } else {
#pragma unroll
            for (int e = 0; e < 8; ++e) {
                yr[idxA[e]] = outR[e];
                yi[idxA[e]] = outI[e];
            }
        }
    }
}

// Full forward 8192-pt complex FFT, in X, result back in X (Y is scratch).
__device__ __forceinline__ void fft8192(
    float* Xr, float* Xi, float* Yr, float* Yi,
    int tid, int wave, int hi, int nc, const Dft16Frag& A)
{
    radix16_wmma_stage<0>(Xr, Xi, Yr, Yi, wave, hi, nc, A);
    __syncthreads();
    radix16_wmma_stage<1>(Yr, Yi, Xr, Xi, wave, hi, nc, A);
    __syncthreads();
    radix16_wmma_stage<2>(Xr, Xi, Yr, Yi, wave, hi, nc, A);
    __syncthreads();
    // final radix-2 (n=2, s=4096), twiddle = 1
    for (int q = tid; q < 4096; q += TPB) {
        const float ar = Yr[q],        ai = Yi[q];
        const float br = Yr[q + 4096], bi = Yi[q + 4096];
        Xr[q]        = ar + br;  Xi[q]        = ai + bi;
        Xr[q + 4096] = ar - br;  Xi[q + 4096] = ai - bi;
    }
    __syncthreads();
}

__global__ __launch_bounds__(TPB)
void pulse_row_kernel(const float* __restrict__ pred,
                      const float* __restrict__ label,
                      float* __restrict__ ws)
{
    extern __shared__ float smem[];
    float* Xr = smem;
    float* Xi = smem + 8192;
    float* Yr = smem + 16384;
    float* Yi = smem + 24576;

    const int b    = blockIdx.x;
    const int tid  = threadIdx.x;
    const int lane = tid & 31;
    const int wave = tid >> 5;
    const int hi   = lane >> 4;
    const int nc   = lane & 15;

    __shared__ int   s_fr, s_lr, s_fi, s_li;
    __shared__ float s_acc;
    if (tid == 0) { s_fr = FFT_N; s_fi = FFT_N; s_lr = -1; s_li = -1; s_acc = 0.f; }

    // ---- warm L2 with this row's labels (64 KB = 512 cachelines); the scan
    //      pass at the end of the serial chain then hits the 192 MB L2 -------
    const float* lab = label + (size_t)b * (2 * FFT_N);
    {
        const char* labc = (const char*)lab;
#pragma unroll
        for (int i = 0; i < 2; ++i)
            __builtin_prefetch(labc + (tid + i * TPB) * 128, 0, 1);
    }

    // Per-lane DFT-16 matrix fragments: A[m][k] = e^{-2*pi*i*m*k/16}
    Dft16Frag A;
#pragma unroll
    for (int kk = 0; kk < 4; ++kk) {
        const int k0 = 4 * kk + 2 * hi;
        float s0, c0, s1, c1;
        __sincosf((float)(nc * k0)       * (-kPI2 / 16.0f), &s0, &c0);
        __sincosf((float)(nc * (k0 + 1)) * (-kPI2 / 16.0f), &s1, &c1);
        A.cosv[kk].x = c0;  A.cosv[kk].y = c1;
        A.sinP[kk].x = s0;  A.sinP[kk].y = s1;
        A.sinN[kk].x = -s0; A.sinN[kk].y = -s1;
    }

    // ---- load prediction row (float4, fully coalesced), imag = 0 ----------
    const float4* p4 = (const float4*)(pred + (size_t)b * FFT_N);
    for (int i = tid; i < FFT_N / 4; i += TPB) {
        const float4 v = p4[i];
        const int j = i * 4;
        Xr[j] = v.x; Xr[j+1] = v.y; Xr[j+2] = v.z; Xr[j+3] = v.w;
        Xi[j] = 0.f; Xi[j+1] = 0.f; Xi[j+2] = 0.f; Xi[j+3] = 0.f;
    }
    __syncthreads();

    // ---- forward FFT -------------------------------------------------------
    fft8192(Xr, Xi, Yr, Yi, tid, wave, hi, nc, A);

    // ---- Hilbert filter H (0 below N/2, 1 at N/2, 2 above) fused with conj
    //      so the inverse FFT is ifft(Z) = conj(fft(conj(Z)))/N --------------
    for (int i = tid; i < FFT_N; i += TPB) {
        const float h = (i < 4096) ? 0.f : ((i == 4096) ? 1.f : 2.f);
        Xr[i] =  h * Xr[i];
        Xi[i] = -h * Xi[i];
    }
    __syncthreads();

    // ---- second FFT; ana = conj(X)/N applied on the fly below --------------
    fft8192(Xr, Xi, Yr, Yi, tid, wave, hi, nc, A);

    // ---- pass A: labels -> LDS (Y buffers free now) + pulse-window scan ----
    int fr = FFT_N, lr = -1, fi = FFT_N, li = -1;
    {
        const float4* lr4 = (const float4*)lab;
        const float4* li4 = (const float4*)(lab + FFT_N);
        for (int i = tid; i < FFT_N / 4; i += TPB) {
            const float4 vr = lr4[i];
            const float4 vi = li4[i];
            const int j = i * 4;
            float rv[4] = {vr.x, vr.y, vr.z, vr.w};
            float iv[4] = {vi.x, vi.y, vi.z, vi.w};
#pragma unroll
            for (int e = 0; e < 4; ++e) {
                Yr[j + e] = rv[e];
                Yi[j + e] = iv[e];
                if (fabsf(rv[e]) > 0.01f) { fr = min(fr, j + e); lr = max(lr, j + e); }
                if (fabsf(iv[e]) > 0.01f) { fi = min(fi, j + e); li = max(li, j + e); }
            }
        }
    }
    atomicMin(&s_fr, fr); atomicMax(&s_lr, lr);
    atomicMin(&s_fi, fi); atomicMax(&s_li, li);
    __syncthreads();

    const int FR = (s_fr == FFT_N) ? 0 : s_fr;       // reference's empty-mask
    const int FI = (s_fi == FFT_N) ? 0 : s_fi;       // defaults (0 / N-1)
    const int LR = (s_lr < 0) ? (FFT_N - 1) : s_lr;
    const int LI = (s_li < 0) ? (FFT_N - 1) : s_li;
    const int first = min(FR, FI);
    const int last  = max(LR, LI);

    // ---- pass B: fused 4-term loss accumulation ----------------------------
    const float invN = 1.0f / (float)FFT_N;
    float acc = 0.f;
    for (int i = tid; i < FFT_N; i += TPB) {
        const float pre =  Xr[i] * invN;             // conj + 1/N of ifft
        const float pim = -Xi[i] * invN;
        const float lre = Yr[i], lim = Yi[i];
        const float dr   = pre - lre;
        const float di   = pim - lim;
        const float dint = (pre * pre + pim * pim) - (lre * lre + lim * lim);
        const float dph  = atan2f(pim, pre) - atan2f(lim, lre);
        const float mask = (i >= first && i < last) ? 1.0f : 0.0f;
        acc += dr * dr + di * di + dint * dint + mask * dph * dph;
    }
    atomicAdd(&s_acc, acc);
    __syncthreads();
    if (tid == 0) {
        // mse_per = acc/(4N); loss contribution = (B-b)*mse_per/B
        ws[b] = (float)(BATCH - b) * s_acc /
                (4.0f * (float)FFT_N * (float)BATCH);
    }
}

__global__ __launch_bounds__(512)
void pulse_reduce_kernel(const float* __restrict__ ws, float* __restrict__ out)
{
    __shared__ float sh[512];
    const int t = threadIdx.x;
    sh[t] = ws[t];
    __syncthreads();
#pragma unroll
    for (int s = 256; s > 0; s >>= 1) {
        if (t < s) sh[t] += sh[t + s];
        __syncthreads();
    }
    if (t == 0) out[0] = sh[0];
}

extern "C" void kernel_launch(void* const* d_in, const int* in_sizes, int n_in,
                              void* d_out, int out_size, void* d_ws, size_t ws_size,
                              hipStream_t stream)
{
    (void)in_sizes; (void)n_in; (void)out_size; (void)ws_size;
    const float* pred  = (const float*)d_in[0];
    const float* label = (const float*)d_in[1];
    // d_in[2] (spectrogram) and d_in[3] (header) are unused by the reference.
    float* ws  = (float*)d_ws;     // 512 per-row weighted losses
    float* out = (float*)d_out;    // scalar

    const size_t shmem = 4u * FFT_N * sizeof(float);  // 128 KB ping-pong
    pulse_row_kernel<<<dim3(BATCH), dim3(TPB), shmem, stream>>>(pred, label, ws);
    pulse_reduce_kernel<<<dim3(1), dim3(512), 0, stream>>>(ws, out);
}